// Backbone_62912680952660
// MI455X (gfx1250) — compile-verified
//
#include <hip/hip_runtime.h>
#include <math.h>

// ---------------- problem constants ----------------
namespace {
constexpr int BB   = 32;
constexpr int H    = 300, W = 190;
constexpr int KH   = 17, KW = 17, SH = 4, SW = 4;
constexpr int nH   = (H - KH) / SH + 1;   // 71
constexpr int nWd  = (W - KW) / SW + 1;   // 44
constexpr int NWIN = nH * nWd;            // 3124
constexpr int NTOT = BB * NWIN;           // 99968
constexpr int L    = 16;                  // quant levels
constexpr int NPX  = KH * KW;             // 289
constexpr int NF   = 40;                  // output features
constexpr int NFP  = 48;                  // padded stride in workspace
constexpr int NWAVES = 8;                 // 256 threads / wave32
}

typedef float v8f_t __attribute__((ext_vector_type(8)));
typedef float v2f_t __attribute__((ext_vector_type(2)));

// ---------------- wave32 cross-lane reductions (no barriers) ----------------
__device__ __forceinline__ float wred_sum(float v) {
#pragma unroll
  for (int m = 16; m > 0; m >>= 1) v += __shfl_xor(v, m, 32);
  return v;
}
__device__ __forceinline__ float wred_min(float v) {
#pragma unroll
  for (int m = 16; m > 0; m >>= 1) v = fminf(v, __shfl_xor(v, m, 32));
  return v;
}
__device__ __forceinline__ float wred_max(float v) {
#pragma unroll
  for (int m = 16; m > 0; m >>= 1) v = fmaxf(v, __shfl_xor(v, m, 32));
  return v;
}

// ---------------- kernel 1: per-window stats + hist + GLCM ----------------
__global__ __launch_bounds__(256)
void glcm_window_kernel(const float* __restrict__ x, const float* __restrict__ bins,
                        float* __restrict__ feats) {
  __shared__ float    s_px[NPX];
  __shared__ int      s_q[NPX];
  __shared__ float    s_bins[15];
  __shared__ unsigned s_cnt[4 * 256];
  __shared__ unsigned s_hist[L];
  __shared__ float    s_w[NWAVES * 8];   // per-wave partials (max 8 quantities)
  __shared__ float    s_res[8];          // combined results
  __shared__ float    s_out[NFP];

  const int tid   = threadIdx.x;
  const int wave  = tid >> 5;
  const int lane  = tid & 31;
  const int idx = blockIdx.x;            // window id in [0, NTOT)
  const int b   = idx / NWIN;
  const int n   = idx - b * NWIN;
  const int wy  = n / nWd;
  const int wx  = n - wy * nWd;
  const int r0  = wy * SH, c0 = wx * SW;
  const float* img = x + (size_t)b * H * W;

  if (tid < 15) s_bins[tid] = bins[tid];
  for (int i = tid; i < 1024; i += 256) s_cnt[i] = 0u;
  if (tid < L) s_hist[tid] = 0u;
  __syncthreads();

  // ---- load + quantize + first-pass stats ----
  float lsum = 0.0f, lmin = 3.0e38f, lmax = -3.0e38f;
  float lqmin = 1000.0f, lqmax = -1000.0f;
  for (int p = tid; p < NPX; p += 256) {
    int r = p / KW, c = p - r * KW;
    float v = img[(size_t)(r0 + r) * W + (c0 + c)];
    s_px[p] = v;
    lsum += v; lmin = fminf(lmin, v); lmax = fmaxf(lmax, v);
    int q = 0;
#pragma unroll
    for (int k = 0; k < 15; ++k) q += (v >= s_bins[k]) ? 1 : 0;
    s_q[p] = q;
    float qf = (float)q;
    lqmin = fminf(lqmin, qf);
    lqmax = fmaxf(lqmax, qf);
  }

  // wave-level reduce 5 stats, combine across 8 waves with one barrier
  {
    float r0v = wred_sum(lsum);
    float r1v = wred_min(lmin);
    float r2v = wred_max(lmax);
    float r3v = wred_min(lqmin);
    float r4v = wred_max(lqmax);
    if (lane == 0) {
      s_w[wave * 8 + 0] = r0v; s_w[wave * 8 + 1] = r1v; s_w[wave * 8 + 2] = r2v;
      s_w[wave * 8 + 3] = r3v; s_w[wave * 8 + 4] = r4v;
    }
    __syncthreads();
    if (tid == 0) {
      float a0 = 0.0f, a1 = 3.0e38f, a2 = -3.0e38f, a3 = 3.0e38f, a4 = -3.0e38f;
#pragma unroll
      for (int w = 0; w < NWAVES; ++w) {
        a0 += s_w[w * 8 + 0];
        a1 = fminf(a1, s_w[w * 8 + 1]);
        a2 = fmaxf(a2, s_w[w * 8 + 2]);
        a3 = fminf(a3, s_w[w * 8 + 3]);
        a4 = fmaxf(a4, s_w[w * 8 + 4]);
      }
      s_res[0] = a0; s_res[1] = a1; s_res[2] = a2; s_res[3] = a3; s_res[4] = a4;
    }
    __syncthreads();
  }
  const float mean = s_res[0] * (1.0f / (float)NPX);
  const float vmin = s_res[1];
  const float vmax = s_res[2];
  const bool  qconst = (s_res[3] == s_res[4]);
  __syncthreads();   // s_w reuse below

  // ---- second-pass variance (population, ddof=0) + start LDS atomics ----
  float lvar = 0.0f;
  for (int p = tid; p < NPX; p += 256) { float d = s_px[p] - mean; lvar += d * d; }
  {
    float rv = wred_sum(lvar);
    if (lane == 0) s_w[wave] = rv;
  }

  // ---- histogram (LDS atomics) ----
  for (int p = tid; p < NPX; p += 256) atomicAdd(&s_hist[s_q[p]], 1u);

  // ---- GLCM pair counting (LDS atomics), offsets (0,1),(1,1),(1,0),(1,-1) ----
  for (int e = tid; e < 17 * 16; e += 256) {           // (0,1)
    int r = e >> 4, c = e & 15;
    int a = s_q[r * KW + c], bb2 = s_q[r * KW + c + 1];
    atomicAdd(&s_cnt[0 * 256 + a * L + bb2], 1u);
  }
  for (int e = tid; e < 16 * 16; e += 256) {           // (1,1)
    int r = e >> 4, c = e & 15;
    int a = s_q[r * KW + c], bb2 = s_q[(r + 1) * KW + c + 1];
    atomicAdd(&s_cnt[1 * 256 + a * L + bb2], 1u);
  }
  for (int e = tid; e < 16 * 17; e += 256) {           // (1,0)
    int r = e / 17, c = e - r * 17;
    int a = s_q[r * KW + c], bb2 = s_q[(r + 1) * KW + c];
    atomicAdd(&s_cnt[2 * 256 + a * L + bb2], 1u);
  }
  for (int e = tid; e < 16 * 16; e += 256) {           // (1,-1)
    int r = e >> 4, c = (e & 15) + 1;
    int a = s_q[r * KW + c], bb2 = s_q[(r + 1) * KW + c - 1];
    atomicAdd(&s_cnt[3 * 256 + a * L + bb2], 1u);
  }
  __syncthreads();   // variance partials, histogram and pair counts all visible

  // ---- stats / hist / padding into s_out ----
  if (tid == 0) {
    float var = 0.0f;
#pragma unroll
    for (int w = 0; w < NWAVES; ++w) var += s_w[w];
    var *= (1.0f / (float)NPX);
    float stdv = sqrtf(var);
    s_out[0] = mean;
    s_out[1] = stdv;
    s_out[2] = (vmax - mean) / stdv;
    s_out[3] = (mean - vmin) / stdv;
  }
  if (tid < L) s_out[4 + tid] = qconst ? 0.0f : (float)s_hist[tid] * (1.0f / (float)NPX);
  if (tid >= NF && tid < NFP) s_out[tid] = 0.0f;
  __syncthreads();   // s_w reuse in GLCM feature phase

  // ---- per-offset GLCM features: 256 threads <-> 256 cells ----
  const float fi = (float)(tid >> 4);
  const float fj = (float)(tid & 15);
  const float d2 = (fi - fj) * (fi - fj);
  const float invd = 1.0f / (1.0f + d2);
  const int ti = tid >> 4, tj = tid & 15;
  const float norms[4] = { 1.0f / (2.0f * 272.0f), 1.0f / (2.0f * 256.0f),
                           1.0f / (2.0f * 272.0f), 1.0f / (2.0f * 256.0f) };

  for (int o = 0; o < 4; ++o) {
    unsigned craw = s_cnt[o * 256 + ti * L + tj] + s_cnt[o * 256 + tj * L + ti]; // symmetrize
    float Pv = (float)craw * norms[o];
    float part[8];
    part[0] = Pv * fi;                  // Si
    part[1] = Pv * fj;                  // Sj
    part[2] = Pv * fi * fi;             // Sii
    part[3] = Pv * fj * fj;             // Sjj
    part[4] = Pv * fi * fj;             // Sij
    part[5] = Pv * invd;                // homogeneity
    part[6] = Pv * Pv;                  // ASM
    part[7] = -Pv * log2f(Pv + 1e-8f);  // entropy

    // wave-level reduce all 8 sums (barrier-free), then one LDS combine
#pragma unroll
    for (int t = 0; t < 8; ++t) part[t] = wred_sum(part[t]);
    if (lane == 0) {
#pragma unroll
      for (int t = 0; t < 8; ++t) s_w[wave * 8 + t] = part[t];
    }
    __syncthreads();
    if (tid == 0) {
      float res[8];
#pragma unroll
      for (int t = 0; t < 8; ++t) {
        float a = 0.0f;
#pragma unroll
        for (int w = 0; w < NWAVES; ++w) a += s_w[w * 8 + t];
        res[t] = a;
      }
      float Si = res[0], Sj = res[1], Sii = res[2], Sjj = res[3], Sij = res[4];
      float contrast = Sii + Sjj - 2.0f * Sij;
      float cov = Sij - Si * Sj;
      float si  = sqrtf(fmaxf(Sii - Si * Si, 0.0f));
      float sj  = sqrtf(fmaxf(Sjj - Sj * Sj, 0.0f));
      float corr = (si < 1e-15f || sj < 1e-15f) ? 1.0f : cov / (si * sj);
      s_out[20 + o] = contrast;
      s_out[24 + o] = res[5];          // homogeneity
      s_out[28 + o] = sqrtf(res[6]);   // energy
      s_out[32 + o] = corr;
      s_out[36 + o] = res[7];          // entropy
    }
    __syncthreads();   // protect s_w for next offset / s_out before store
  }

  if (tid < NFP) feats[(size_t)idx * NFP + tid] = s_out[tid];
}

// ---------------- kernel 2: mean over windows via V_WMMA_F32_16X16X4_F32 ----------------
// D = ones(16x4) x B(4x16) + C accumulated over 781 K-chunks => every row of D
// holds the sum over all 3124 windows of the 16 features in this tile (exact f32 sums).
__global__ __launch_bounds__(32)
void reduce_mean_wmma(const float* __restrict__ feats, float* __restrict__ out) {
  const int b    = blockIdx.x;   // batch element
  const int tile = blockIdx.y;   // feature tile: 0..2 (16 features each, tile 2 padded)
  const int lane = threadIdx.x;  // 0..31, full wave, EXEC all-ones through WMMA
  const int col   = lane & 15;
  const int khalf = lane >> 4;   // K-interleave: VGPR0 holds K = 2*khalf, VGPR1 holds K = 2*khalf+1

  v2f_t a; a[0] = 1.0f; a[1] = 1.0f;   // all-ones A: lane layout irrelevant
  v8f_t c = {};

  const float* base = feats + (size_t)b * NWIN * NFP + tile * 16 + col;
  for (int k0 = 0; k0 < NWIN; k0 += 4) {   // NWIN = 3124 = 781*4, no tail
    v2f_t bm;
    bm[0] = base[(size_t)(k0 + 2 * khalf)     * NFP];
    bm[1] = base[(size_t)(k0 + 2 * khalf + 1) * NFP];
    c = __builtin_amdgcn_wmma_f32_16x16x4_f32(
        /*neg_a=*/false, a, /*neg_b=*/false, bm,
        /*c_mod=*/(short)0, c, /*reuse_a=*/false, /*reuse_b=*/false);
  }

  // Row M=0 of D lives in accumulator VGPR0, lanes 0-15 (cols 0-15).
  float s = c[0];
  int f = tile * 16 + col;
  if (lane < 16 && f < NF)
    out[(size_t)b * NF + f] = s * (1.0f / (float)NWIN);
}

// ---------------- host entry ----------------
extern "C" void kernel_launch(void* const* d_in, const int* in_sizes, int n_in,
                              void* d_out, int out_size, void* d_ws, size_t ws_size,
                              hipStream_t stream) {
  (void)in_sizes; (void)n_in; (void)out_size; (void)ws_size;
  const float* x    = (const float*)d_in[0];   // (32,1,300,190) f32
  const float* bins = (const float*)d_in[1];   // (15,) f32
  float* out   = (float*)d_out;                // (32,40) f32
  float* feats = (float*)d_ws;                 // NTOT x 48 f32 scratch (~19.2 MB)

  glcm_window_kernel<<<NTOT, 256, 0, stream>>>(x, bins, feats);

  dim3 g2(BB, 3);
  reduce_mean_wmma<<<g2, 32, 0, stream>>>(feats, out);
}